// FALayer_16956530884761
// MI455X (gfx1250) — compile-verified
//
#include <hip/hip_runtime.h>

#define DIM 128
#define OUT_LD 256          // NUM_GRAPH * DIM, output row stride (compile-time)
#define BETA 0.3f

typedef __attribute__((ext_vector_type(2))) float v2f;
typedef __attribute__((ext_vector_type(8))) float v8f;

// ---------------------------------------------------------------------------
// pd[i] = dot(h[i,:], gw[0:128]); ps[i] = dot(h[i,:], gw[128:256])
// One wave32 per node; each lane owns 4 contiguous floats (float4 load).
// ---------------------------------------------------------------------------
__global__ void __launch_bounds__(256)
gate_kernel(const float* __restrict__ h, const float* __restrict__ gw,
            float* __restrict__ pd, float* __restrict__ ps, int n) {
    int node = (int)((blockIdx.x * blockDim.x + threadIdx.x) >> 5);
    int lane = threadIdx.x & 31;
    if (node >= n) return;
    float4 hv = ((const float4*)(h + (size_t)node * DIM))[lane];
    float4 wd = ((const float4*)gw)[lane];
    float4 wsv = ((const float4*)(gw + DIM))[lane];
    float a = hv.x * wd.x + hv.y * wd.y + hv.z * wd.z + hv.w * wd.w;
    float b = hv.x * wsv.x + hv.y * wsv.y + hv.z * wsv.z + hv.w * wsv.w;
#pragma unroll
    for (int off = 16; off > 0; off >>= 1) {
        a += __shfl_xor(a, off, 32);
        b += __shfl_xor(b, off, 32);
    }
    if (lane == 0) { pd[node] = a; ps[node] = b; }
}

// ---------------------------------------------------------------------------
// Zero the aggregation buffer (float4 granularity).
// ---------------------------------------------------------------------------
__global__ void __launch_bounds__(256)
zero_kernel(float4* __restrict__ p, int n4) {
    int i = blockIdx.x * blockDim.x + threadIdx.x;
    if (i < n4) p[i] = make_float4(0.f, 0.f, 0.f, 0.f);
}

// ---------------------------------------------------------------------------
// Per-edge: gate = tanh(pd[dst] + ps[src] + gb); agg[dst,:] += h[src,:]*gate.
// One wave32 per edge; float4 gather per lane + 4 f32 atomics per lane.
// h/agg are L2-resident (25.6 MB each << 192 MB L2), so the random
// gather/scatter never touches HBM after first use.
// ---------------------------------------------------------------------------
__global__ void __launch_bounds__(256)
edge_kernel(const float* __restrict__ h,
            const int* __restrict__ src, const int* __restrict__ dst,
            const float* __restrict__ pd, const float* __restrict__ ps,
            const float* __restrict__ gate_b, int gb_idx,
            float* __restrict__ agg, int E) {
    int e = (int)((blockIdx.x * blockDim.x + threadIdx.x) >> 5);
    int lane = threadIdx.x & 31;
    if (e >= E) return;
    int s = src[e];
    int d = dst[e];
    float g = tanhf(pd[d] + ps[s] + gate_b[gb_idx]);
    float4 hv = ((const float4*)(h + (size_t)s * DIM))[lane];
    float* ap = agg + (size_t)d * DIM + lane * 4;
    atomicAdd(ap + 0, hv.x * g);
    atomicAdd(ap + 1, hv.y * g);
    atomicAdd(ap + 2, hv.z * g);
    atomicAdd(ap + 3, hv.w * g);
}

// ---------------------------------------------------------------------------
// h = agg*(1-BETA) + x*BETA   (elementwise, float4)
// ---------------------------------------------------------------------------
__global__ void __launch_bounds__(256)
blend_kernel(const float4* __restrict__ agg, const float4* __restrict__ x,
             float4* __restrict__ h, int n4) {
    int i = blockIdx.x * blockDim.x + threadIdx.x;
    if (i >= n4) return;
    float4 a = agg[i];
    float4 xv = x[i];
    float4 r;
    r.x = a.x * (1.0f - BETA) + xv.x * BETA;
    r.y = a.y * (1.0f - BETA) + xv.y * BETA;
    r.z = a.z * (1.0f - BETA) + xv.z * BETA;
    r.w = a.w * (1.0f - BETA) + xv.w * BETA;
    h[i] = r;
}

// ---------------------------------------------------------------------------
// out[:, col0:col0+128] = relu(h @ W + bias) via V_WMMA_F32_16X16X4_F32.
// Block = 256 threads = 8 waves; block owns a 16-row strip of h (staged in
// LDS, 8 KB); wave w computes the 16x16 tile at columns [16w, 16w+16).
// K loop: 128/4 = 32 WMMAs per wave, unrolled x8 so B loads clause deeply.
// Epilogue: compile-time OUT_LD -> 8 immediate-offset stores, no per-element
// exec-mask branches on the (always-taken for N=50000) full-tile path.
// ---------------------------------------------------------------------------
__global__ void __launch_bounds__(256)
out_gemm_kernel(const float* __restrict__ h, const float* __restrict__ W,
                const float* __restrict__ bias, float* __restrict__ out,
                int n, int col0) {
    __shared__ float As[16 * DIM];  // 8 KB: 16 rows x 128 cols of h

    int row0 = blockIdx.x * 16;
    int tid = threadIdx.x;

    // Cooperative load of the A strip: 512 float4s across 256 threads.
#pragma unroll
    for (int i = tid; i < 512; i += 256) {
        int r = i >> 5;  // 32 float4 per row
        int gr = row0 + r;
        float4 v = (gr < n) ? ((const float4*)(h + (size_t)gr * DIM))[i & 31]
                            : make_float4(0.f, 0.f, 0.f, 0.f);
        ((float4*)As)[i] = v;
    }
    __syncthreads();

    int wave = tid >> 5;         // 0..7 -> output column tile
    int lane = tid & 31;
    int n0 = wave * 16;
    int m = lane & 15;           // row within A tile / col within B,C tile
    int koff = (lane >> 4) * 2;  // lanes 16-31 carry K+2,K+3

    v8f c = {0.f, 0.f, 0.f, 0.f, 0.f, 0.f, 0.f, 0.f};
#pragma unroll 8
    for (int k0 = 0; k0 < DIM; k0 += 4) {
        // A frag (16x4 f32): lane m holds A[m][k0+koff], A[m][k0+koff+1]
        const float* ap = As + m * DIM + k0 + koff;
        v2f a;
        a.x = ap[0];
        a.y = ap[1];
        // B frag (4x16 f32), W row-major [K=128][N=128]
        const float* bp = W + (size_t)(k0 + koff) * DIM + n0 + m;
        v2f b;
        b.x = bp[0];
        b.y = bp[DIM];
        c = __builtin_amdgcn_wmma_f32_16x16x4_f32(
                /*neg_a=*/false, a, /*neg_b=*/false, b,
                /*c_mod=*/(short)0, c, /*reuse_a=*/false, /*reuse_b=*/false);
    }

    // C/D layout: VGPR v -> M = v + 8*(lane>=16); N = lane&15.
    int nc = n0 + m;
    float bv = bias[nc];
    int mbase = row0 + ((lane >> 4) << 3);
    float* op = out + (size_t)mbase * OUT_LD + col0 + nc;
    if (mbase + 8 <= n) {
        // Full tile: 8 stores at constant 1024B stride -> immediate offsets.
#pragma unroll
        for (int v = 0; v < 8; ++v) {
            float val = c[v] + bv;
            op[v * OUT_LD] = val > 0.f ? val : 0.f;
        }
    } else {
#pragma unroll
        for (int v = 0; v < 8; ++v) {
            if (mbase + v < n) {
                float val = c[v] + bv;
                op[v * OUT_LD] = val > 0.f ? val : 0.f;
            }
        }
    }
}

// ---------------------------------------------------------------------------
// Orchestration. Inputs (setup_inputs order):
//   0: x      [N*128] f32      1: src [E] i32       2: dst [E] i32
//   3: gate_w [4*256] f32      4: gate_b [4] f32
//   5: out_w  [2*128*128] f32  6: out_b [2*128] f32
// Output: [N, 256] f32 (graph 0 cols 0..127, graph 1 cols 128..255).
// ---------------------------------------------------------------------------
extern "C" void kernel_launch(void* const* d_in, const int* in_sizes, int n_in,
                              void* d_out, int out_size, void* d_ws, size_t ws_size,
                              hipStream_t stream) {
    const float* x = (const float*)d_in[0];
    const int* src = (const int*)d_in[1];
    const int* dst = (const int*)d_in[2];
    const float* gate_w = (const float*)d_in[3];
    const float* gate_b = (const float*)d_in[4];
    const float* out_w = (const float*)d_in[5];
    const float* out_b = (const float*)d_in[6];
    float* out = (float*)d_out;

    int N = in_sizes[0] / DIM;
    int E = in_sizes[1];
    size_t nd = (size_t)N * DIM;
    int n4 = (int)(nd / 4);

    float* h = (float*)d_ws;        // N*128
    float* agg = h + nd;            // N*128
    float* pd = agg + nd;           // N
    float* ps = pd + N;             // N

    int eltBlocks = (n4 + 255) / 256;
    int nodeWaveBlocks = (N + 7) / 8;    // 8 waves (nodes) per 256-thr block
    int edgeWaveBlocks = (E + 7) / 8;    // 8 waves (edges) per 256-thr block
    int gemmBlocks = (N + 15) / 16;

    for (int g = 0; g < 2; ++g) {
        for (int k = 0; k < 2; ++k) {
            const float* hin = (k == 0) ? x : h;
            int idx = g * 2 + k;
            gate_kernel<<<nodeWaveBlocks, 256, 0, stream>>>(
                hin, gate_w + (size_t)idx * 2 * DIM, pd, ps, N);
            zero_kernel<<<eltBlocks, 256, 0, stream>>>((float4*)agg, n4);
            edge_kernel<<<edgeWaveBlocks, 256, 0, stream>>>(
                hin, src, dst, pd, ps, gate_b, idx, agg, E);
            blend_kernel<<<eltBlocks, 256, 0, stream>>>(
                (const float4*)agg, (const float4*)x, (float4*)h, n4);
        }
        out_gemm_kernel<<<gemmBlocks, 256, 0, stream>>>(
            h, out_w + (size_t)g * DIM * DIM, out_b + (size_t)g * DIM,
            out, N, g * DIM);
    }
}